// GCN_LSTM_OD_45140106281284
// MI455X (gfx1250) — compile-verified
//
#include <hip/hip_runtime.h>
#include <math.h>

// ---------------- problem constants ----------------
static constexpr int T_STEPS = 24;
static constexpr int B_BATCH = 8;
static constexpr int N_NODES = 512;
static constexpr int M_NODES = 4096;   // B*N flattened nodes
static constexpr int HID     = 64;
static constexpr int E_EDGES = 16384;
static constexpr int PRED    = 6;
static constexpr int CHUNKS  = 8;      // row chunks per (b,t) slice in feat pass

typedef __attribute__((ext_vector_type(2))) float v2f;
typedef __attribute__((ext_vector_type(8))) float v8f;

// ---------------- workspace layout (floats) ----------------
static constexpr size_t FEAT_OFF = 0;                                   // T*M*2
static constexpr size_t DIS_OFF  = FEAT_OFF + (size_t)T_STEPS*M_NODES*2;
static constexpr size_t SELF_OFF = DIS_OFF  + M_NODES;
static constexpr size_t CNRM_OFF = SELF_OFF + M_NODES;                  // E
static constexpr size_t H1_OFF   = CNRM_OFF + E_EDGES;                  // M*64
static constexpr size_t HW_OFF   = H1_OFF   + (size_t)M_NODES*HID;
static constexpr size_t XT_OFF   = HW_OFF   + (size_t)M_NODES*HID;
static constexpr size_t H_OFF    = XT_OFF   + (size_t)M_NODES*HID;
static constexpr size_t C_OFF    = H_OFF    + (size_t)M_NODES*HID;      // must follow H (zeroed together)
static constexpr size_t G_OFF    = C_OFF    + (size_t)M_NODES*HID;      // M*256
static constexpr size_t HO_OFF   = G_OFF    + (size_t)M_NODES*4*HID;
static constexpr size_t HD_OFF   = HO_OFF   + (size_t)M_NODES*HID;
static constexpr size_t COLP_OFF = HD_OFF   + (size_t)M_NODES*HID;      // B*T*CHUNKS*N
static constexpr size_t FLT_END  = COLP_OFF + (size_t)B_BATCH*T_STEPS*CHUNKS*N_NODES;
// int region (starts at FLT_END floats)
static constexpr size_t CNT_IOFF  = 0;                  // M ints
static constexpr size_t OFFS_IOFF = CNT_IOFF + M_NODES; // 513 ints
static constexpr size_t CSRC_IOFF = OFFS_IOFF + 520;    // E ints

// ---------------- feat pass A: per-chunk row sums + column partials -------
// grid = B*T*CHUNKS blocks, 256 threads (8 waves). Block owns 64 rows.
__global__ void feat_rows_kernel(const float* __restrict__ xlog, float* __restrict__ feat,
                                 float* __restrict__ colp) {
    __shared__ float wcols[8 * N_NODES];   // 16 KB
    const int blk   = blockIdx.x;
    const int chunk = blk & (CHUNKS - 1);
    const int bt    = blk >> 3;            // b*T + t
    const int b = bt / T_STEPS, t = bt % T_STEPS;
    const int wave = threadIdx.x >> 5, lane = threadIdx.x & 31;
    const float* base = xlog + (size_t)bt * N_NODES * N_NODES
                             + (size_t)chunk * 64 * N_NODES;

    float4 cacc[4];
#pragma unroll
    for (int k = 0; k < 4; ++k) cacc[k] = make_float4(0.f, 0.f, 0.f, 0.f);

#pragma unroll
    for (int it = 0; it < 8; ++it) {
        int lr = wave + 8 * it;            // local row 0..63
        const float* rowp = base + (size_t)lr * N_NODES;
        float rsum = 0.f;
#pragma unroll
        for (int k = 0; k < 4; ++k) {
            float4 v = *(const float4*)(rowp + lane * 4 + 128 * k);
            v.x = expm1f(v.x); v.y = expm1f(v.y);
            v.z = expm1f(v.z); v.w = expm1f(v.w);
            rsum += (v.x + v.y) + (v.z + v.w);
            cacc[k].x += v.x; cacc[k].y += v.y;
            cacc[k].z += v.z; cacc[k].w += v.w;
        }
#pragma unroll
        for (int o = 16; o > 0; o >>= 1) rsum += __shfl_xor(rsum, o, 32);
        if (lane == 0) {
            int r = chunk * 64 + lr;
            feat[((size_t)t * M_NODES + (size_t)b * N_NODES + r) * 2 + 0] = rsum;
        }
    }
#pragma unroll
    for (int k = 0; k < 4; ++k) {
        int c = lane * 4 + 128 * k;
        wcols[wave * N_NODES + c + 0] = cacc[k].x;
        wcols[wave * N_NODES + c + 1] = cacc[k].y;
        wcols[wave * N_NODES + c + 2] = cacc[k].z;
        wcols[wave * N_NODES + c + 3] = cacc[k].w;
    }
    __syncthreads();
    for (int c = threadIdx.x; c < N_NODES; c += 256) {
        float s = 0.f;
#pragma unroll
        for (int w = 0; w < 8; ++w) s += wcols[w * N_NODES + c];
        colp[(size_t)blk * N_NODES + c] = s;   // blk = bt*CHUNKS + chunk
    }
}

// ---------------- feat pass B: reduce column partials (fixed order) -------
__global__ void feat_cols_kernel(const float* __restrict__ colp, float* __restrict__ feat) {
    int gid = blockIdx.x * blockDim.x + threadIdx.x;   // B*T*N
    if (gid >= B_BATCH * T_STEPS * N_NODES) return;
    int c  = gid & (N_NODES - 1);
    int bt = gid >> 9;
    int b = bt / T_STEPS, t = bt % T_STEPS;
    float s = 0.f;
#pragma unroll
    for (int k = 0; k < CHUNKS; ++k)
        s += colp[((size_t)bt * CHUNKS + k) * N_NODES + c];
    feat[((size_t)t * M_NODES + (size_t)b * N_NODES + c) * 2 + 1] = s;
}

// ---------------- degree / csr helpers ----------------
__global__ void zero_int_kernel(int* p, int n) {
    int i = blockIdx.x * blockDim.x + threadIdx.x;
    if (i < n) p[i] = 0;
}
__global__ void zero_f_kernel(float* p, int n) {
    int i = blockIdx.x * blockDim.x + threadIdx.x;
    if (i < n) p[i] = 0.f;
}
__global__ void count_kernel(const int* __restrict__ dst, int* __restrict__ cnt) {
    int e = blockIdx.x * blockDim.x + threadIdx.x;
    if (e < E_EDGES) {
        int d = dst[e];
        if (d >= 0 && d < M_NODES) atomicAdd(&cnt[d], 1);
    }
}
__global__ void norm_kernel(const int* __restrict__ cnt, float* __restrict__ dis,
                            float* __restrict__ selfn) {
    int m = blockIdx.x * blockDim.x + threadIdx.x;
    if (m < M_NODES) {
        float deg = 1.f + (float)cnt[m];
        dis[m]   = rsqrtf(deg);
        selfn[m] = 1.f / deg;
    }
}
// single block, 512 threads: prefix-sum counts, deterministic ordered fill
__global__ void csr_kernel(const int* __restrict__ src, const int* __restrict__ dst,
                           const int* __restrict__ cnt, const float* __restrict__ dis,
                           int* __restrict__ offs, int* __restrict__ csr_src,
                           float* __restrict__ csr_norm) {
    __shared__ int s[N_NODES];
    int tid = threadIdx.x;
    s[tid] = cnt[tid];
    __syncthreads();
    for (int off = 1; off < N_NODES; off <<= 1) {
        int v = s[tid];
        if (tid >= off) v += s[tid - off];
        __syncthreads();
        s[tid] = v;
        __syncthreads();
    }
    if (tid == 0) offs[0] = 0;
    offs[tid + 1] = s[tid];
    int pos = (tid == 0) ? 0 : s[tid - 1];   // exclusive start
    float dd = dis[tid];
    for (int e = 0; e < E_EDGES; ++e) {
        if (dst[e] == tid) {
            int sv = src[e];
            csr_src[pos]  = sv;
            csr_norm[pos] = dis[sv] * dd;
            ++pos;
        }
    }
}

// ---------------- GCN layer 1 (in_dim=2 -> 64, fused conv+agg+relu) -------
__global__ void gcn1_kernel(const float* __restrict__ feat_t, const float* __restrict__ W1,
                            const float* __restrict__ b1, const float* __restrict__ selfn,
                            const int* __restrict__ offs, const int* __restrict__ csr_src,
                            const float* __restrict__ csr_norm, float* __restrict__ h1) {
    int gid = blockIdx.x * blockDim.x + threadIdx.x;   // M*64
    int r = gid >> 6, f = gid & 63;
    float w0 = W1[f], w1 = W1[HID + f];
    float val = (feat_t[r * 2] * w0 + feat_t[r * 2 + 1] * w1) * selfn[r] + b1[f];
    if (r < N_NODES) {
        int e1 = offs[r + 1];
        for (int e = offs[r]; e < e1; ++e) {
            int sv = csr_src[e];
            val += csr_norm[e] * (feat_t[sv * 2] * w0 + feat_t[sv * 2 + 1] * w1);
        }
    }
    h1[gid] = fmaxf(val, 0.f);
}

// ---------------- GCN aggregation (layer 2 second half) -------------------
__global__ void gcn_agg_kernel(const float* __restrict__ hw, const float* __restrict__ bias,
                               const float* __restrict__ selfn, const int* __restrict__ offs,
                               const int* __restrict__ csr_src, const float* __restrict__ csr_norm,
                               float* __restrict__ outp) {
    int gid = blockIdx.x * blockDim.x + threadIdx.x;   // M*64
    int r = gid >> 6, f = gid & 63;
    float val = hw[gid] * selfn[r] + bias[f];
    if (r < N_NODES) {
        int e1 = offs[r + 1];
        for (int e = offs[r]; e < e1; ++e)
            val += csr_norm[e] * hw[csr_src[e] * HID + f];
    }
    outp[gid] = fmaxf(val, 0.f);
}

// ---------------- f32 WMMA GEMM (K=64): C = (ACCUM? C:0) + A(Mx64) * B ----
// BT==0: B element (k,n) = Bm[k*Nn + n];  BT==1: (k,n) = Bm[n*64 + k]  (A @ W^T)
// one wave per 16x16 C tile; straight-line unrolled V_WMMA_F32_16X16X4_F32.
template <int BT, int ACCUM>
__global__ void gemm_wmma(const float* __restrict__ A, const float* __restrict__ Bm,
                          const float* __restrict__ bias, float* __restrict__ C,
                          int Mrows, int Nn) {
    constexpr int K = HID;   // 64 for every GEMM in this model
    int wid  = blockIdx.x * (blockDim.x >> 5) + (threadIdx.x >> 5);
    int lane = threadIdx.x & 31;
    int tilesN = Nn >> 4;
    int tiles  = (Mrows >> 4) * tilesN;
    if (wid >= tiles) return;
    int tile_n = wid % tilesN;
    int tile_m = wid / tilesN;
    int half = lane >> 4;
    int lm   = lane & 15;
    int rowA = tile_m * 16 + lm;
    int col  = tile_n * 16 + lm;

    v8f acc = {0.f, 0.f, 0.f, 0.f, 0.f, 0.f, 0.f, 0.f};
    if (ACCUM) {
#pragma unroll
        for (int j = 0; j < 8; ++j)
            acc[j] = C[(size_t)(tile_m * 16 + j + 8 * half) * Nn + col];
    }
#pragma unroll
    for (int k0 = 0; k0 < K; k0 += 4) {
        int ka = k0 + half * 2;
        v2f a, b;
        a.x = A[(size_t)rowA * K + ka];
        a.y = A[(size_t)rowA * K + ka + 1];
        if (BT) {
            b.x = Bm[(size_t)col * K + ka];
            b.y = Bm[(size_t)col * K + ka + 1];
        } else {
            b.x = Bm[(size_t)ka * Nn + col];
            b.y = Bm[(size_t)(ka + 1) * Nn + col];
        }
        acc = __builtin_amdgcn_wmma_f32_16x16x4_f32(false, a, false, b,
                                                    (short)0, acc, false, false);
    }
    float bv = bias ? bias[col] : 0.f;
#pragma unroll
    for (int j = 0; j < 8; ++j) {
        int row = tile_m * 16 + j + 8 * half;
        C[(size_t)row * Nn + col] = acc[j] + bv;
    }
}

// ---------------- LSTM gate update ----------------------------------------
__global__ void lstm_gates_kernel(const float* __restrict__ G, const float* __restrict__ bih,
                                  const float* __restrict__ bhh, float* __restrict__ h,
                                  float* __restrict__ c) {
    int gid = blockIdx.x * blockDim.x + threadIdx.x;   // M*64
    int m = gid >> 6, j = gid & 63;
    const float* g = G + (size_t)m * 4 * HID;
    float gi = g[j]           + bih[j]           + bhh[j];
    float gf = g[HID + j]     + bih[HID + j]     + bhh[HID + j];
    float gg = g[2 * HID + j] + bih[2 * HID + j] + bhh[2 * HID + j];
    float go = g[3 * HID + j] + bih[3 * HID + j] + bhh[3 * HID + j];
    float si = 1.f / (1.f + expf(-gi));
    float sf = 1.f / (1.f + expf(-gf));
    float so = 1.f / (1.f + expf(-go));
    float cn = sf * c[gid] + si * tanhf(gg);
    c[gid] = cn;
    h[gid] = so * tanhf(cn);
}

// ---------------- batched outer product + log1p(relu) + 6 copies ----------
// per-batch 512x512 = 32x32 tiles of 16x16; 8192 waves total.
__global__ void od_kernel(const float* __restrict__ Ho, const float* __restrict__ Hd,
                          float* __restrict__ outp) {
    int wid  = blockIdx.x * (blockDim.x >> 5) + (threadIdx.x >> 5);
    int lane = threadIdx.x & 31;
    if (wid >= B_BATCH * 1024) return;
    int b  = wid >> 10;
    int tl = wid & 1023;
    int tn = tl >> 5;      // row tile (n)
    int tm = tl & 31;      // col tile (m)
    int half = lane >> 4;
    int lm   = lane & 15;
    const float* Arow = Ho + (size_t)(b * N_NODES + tn * 16 + lm) * HID;
    const float* Brow = Hd + (size_t)(b * N_NODES + tm * 16 + lm) * HID;

    v8f acc = {0.f, 0.f, 0.f, 0.f, 0.f, 0.f, 0.f, 0.f};
#pragma unroll
    for (int k0 = 0; k0 < HID; k0 += 4) {
        int ka = k0 + half * 2;
        v2f a, bb;
        a.x  = Arow[ka];
        a.y  = Arow[ka + 1];
        bb.x = Brow[ka];
        bb.y = Brow[ka + 1];
        acc = __builtin_amdgcn_wmma_f32_16x16x4_f32(false, a, false, bb,
                                                    (short)0, acc, false, false);
    }
#pragma unroll
    for (int j = 0; j < 8; ++j) {
        int n = tn * 16 + j + 8 * half;
        int m = tm * 16 + lm;
        float v = log1pf(fmaxf(acc[j], 0.f));
        size_t base = ((size_t)b * PRED) * N_NODES * N_NODES + (size_t)n * N_NODES + m;
#pragma unroll
        for (int p = 0; p < PRED; ++p)
            outp[base + (size_t)p * N_NODES * N_NODES] = v;
    }
}

// ---------------- host orchestration --------------------------------------
extern "C" void kernel_launch(void* const* d_in, const int* in_sizes, int n_in,
                              void* d_out, int out_size, void* d_ws, size_t ws_size,
                              hipStream_t stream) {
    const float* xlog = (const float*)d_in[0];
    const int*   eidx = (const int*)d_in[1];        // [src(E), dst(E)]
    const float* W1 = (const float*)d_in[2];
    const float* b1 = (const float*)d_in[3];
    const float* W2 = (const float*)d_in[4];
    const float* b2 = (const float*)d_in[5];
    const float* Wih = (const float*)d_in[6];
    const float* Whh = (const float*)d_in[7];
    const float* bih = (const float*)d_in[8];
    const float* bhh = (const float*)d_in[9];
    const float* Wo = (const float*)d_in[10];
    const float* bo = (const float*)d_in[11];
    const float* Wd = (const float*)d_in[12];
    const float* bd = (const float*)d_in[13];
    float* outp = (float*)d_out;

    float* fws   = (float*)d_ws;
    float* feat  = fws + FEAT_OFF;
    float* dis   = fws + DIS_OFF;
    float* selfn = fws + SELF_OFF;
    float* cnorm = fws + CNRM_OFF;
    float* h1    = fws + H1_OFF;
    float* hw    = fws + HW_OFF;
    float* xt    = fws + XT_OFF;
    float* hstate= fws + H_OFF;
    float* cstate= fws + C_OFF;
    float* G     = fws + G_OFF;
    float* Ho    = fws + HO_OFF;
    float* Hd    = fws + HD_OFF;
    float* colp  = fws + COLP_OFF;
    int* iws   = (int*)(fws + FLT_END);
    int* cnt   = iws + CNT_IOFF;
    int* offs  = iws + OFFS_IOFF;
    int* csrc  = iws + CSRC_IOFF;

    const int* e_src = eidx;
    const int* e_dst = eidx + E_EDGES;

    // 1) features from the big OD tensor (memory-bound pass, split for occupancy)
    feat_rows_kernel<<<B_BATCH * T_STEPS * CHUNKS, 256, 0, stream>>>(xlog, feat, colp);
    feat_cols_kernel<<<(B_BATCH * T_STEPS * N_NODES) / 256, 256, 0, stream>>>(colp, feat);

    // 2) graph normalization + deterministic CSR
    zero_int_kernel<<<M_NODES / 256, 256, 0, stream>>>(cnt, M_NODES);
    count_kernel<<<E_EDGES / 256, 256, 0, stream>>>(e_dst, cnt);
    norm_kernel<<<M_NODES / 256, 256, 0, stream>>>(cnt, dis, selfn);
    csr_kernel<<<1, N_NODES, 0, stream>>>(e_src, e_dst, cnt, dis, offs, csrc, cnorm);

    // 3) zero LSTM state (h and c contiguous)
    zero_f_kernel<<<(2 * M_NODES * HID) / 256, 256, 0, stream>>>(hstate, 2 * M_NODES * HID);

    // 4) per-timestep GCN encode + LSTM step (sequential dependency)
    for (int t = 0; t < T_STEPS; ++t) {
        const float* feat_t = feat + (size_t)t * M_NODES * 2;
        gcn1_kernel<<<(M_NODES * HID) / 256, 256, 0, stream>>>(
            feat_t, W1, b1, selfn, offs, csrc, cnorm, h1);
        // hw = h1 @ W2   (4096x64 * 64x64) — WMMA
        gemm_wmma<0, 0><<<256, 128, 0, stream>>>(h1, W2, nullptr, hw, M_NODES, HID);
        gcn_agg_kernel<<<(M_NODES * HID) / 256, 256, 0, stream>>>(
            hw, b2, selfn, offs, csrc, cnorm, xt);
        // G = xt @ Wih^T ; G += h @ Whh^T   (4096x256) — WMMA
        gemm_wmma<1, 0><<<1024, 128, 0, stream>>>(xt, Wih, nullptr, G, M_NODES, 4 * HID);
        gemm_wmma<1, 1><<<1024, 128, 0, stream>>>(hstate, Whh, nullptr, G, M_NODES, 4 * HID);
        lstm_gates_kernel<<<(M_NODES * HID) / 256, 256, 0, stream>>>(G, bih, bhh, hstate, cstate);
    }

    // 5) output projections + batched outer product
    gemm_wmma<1, 0><<<256, 128, 0, stream>>>(hstate, Wo, bo, Ho, M_NODES, HID);
    gemm_wmma<1, 0><<<256, 128, 0, stream>>>(hstate, Wd, bd, Hd, M_NODES, HID);
    od_kernel<<<2048, 128, 0, stream>>>(Ho, Hd, outp);
}